// MultiHeadAttention_21869973471801
// MI455X (gfx1250) — compile-verified
//
#include <hip/hip_runtime.h>
#include <hip/hip_bf16.h>

// ---------------------------------------------------------------------------
// MI455X (gfx1250) fused multi-head-attention pipeline, fp32 end-to-end using
// V_WMMA_F32_16X16X4_F32 for every matmul (exact fp32 matrix path).
//   b=4, s=2048, d=1024, h=16, hd=64  (from reference setup_inputs()).
//
// LDS staging uses "fragment order" so every WMMA operand fragment is a
// single aligned ds_load_b64 into an even VGPR pair (no v_mov repacking):
//   frag dword addr = base + half*32 + 2*l16 (+sub)  -> banks 0..63 exactly.
//
// Tile staging uses gfx1250 ASYNC global->LDS copies (ASYNCcnt-tracked):
// each lane carries its own LDS destination, so the frag-order scatter is
// done by the async unit directly -- no VGPR round-trip, no loadcnt stalls.
// ---------------------------------------------------------------------------

typedef __attribute__((ext_vector_type(2))) float v2f;
typedef __attribute__((ext_vector_type(8))) float v8f;

#define LOG2_E 1.442695041f
#define RMS_EPS 1e-6f

// A-fragment storage for a (mtiles x 64) fp32 chunk:
//   element (m = mt*16 + ml, k = kk*4 + half*2 + sub)
#define AFRAG(mt, kk, half, l16) ((((mt) * 16 + (kk)) * 64) + (half) * 32 + (l16) * 2)
// B-fragment storage for a (64 x 64) fp32 chunk (kk k-steps x 4 n-tiles):
//   element (k = kk*4 + half*2 + sub, n = j*16 + l16)
#define BFRAG(kk, j, half, l16) ((((kk) * 4 + (j)) * 64) + (half) * 32 + (l16) * 2)

#define GSTRIDE 68   // padded row stride for row-major K tile (bank = 4*m + c)

// LDS offset of a shared-memory pointer: generic LDS addresses carry the
// wave-relative LDS offset in ADDR[31:0] (shared aperture in the high bits).
__device__ __forceinline__ unsigned lds_off(const void* p) {
    return (unsigned)(size_t)p;
}
// Async global->LDS copies (gfx1250): copy from the global address in src to the LDS address in ldst, ASYNCcnt-tracked.
__device__ __forceinline__ void async_ld_b32(unsigned ldst, const void* src) {
    asm volatile("global_load_async_to_lds_b32 %0, %1, off"
                 :: "v"(ldst), "v"(src) : "memory");
}
__device__ __forceinline__ void async_ld_b64(unsigned ldst, const void* src) {
    asm volatile("global_load_async_to_lds_b64 %0, %1, off"
                 :: "v"(ldst), "v"(src) : "memory");
}
__device__ __forceinline__ void async_ld_b128(unsigned ldst, const void* src) {
    asm volatile("global_load_async_to_lds_b128 %0, %1, off"
                 :: "v"(ldst), "v"(src) : "memory");
}
__device__ __forceinline__ void wait_async0() {
    asm volatile("s_wait_asynccnt 0x0" ::: "memory");
}

// ---------------------------------------------------------------------------
// Kernel 1: num_masked[b] = sum over s of patch_mask[b, s]
// ---------------------------------------------------------------------------
__global__ void mask_sum_kernel(const unsigned char* __restrict__ mask,
                                int* __restrict__ out, int S) {
    __shared__ int red[256];
    const int bb = blockIdx.x;
    int acc = 0;
    for (int i = threadIdx.x; i < S; i += 256)
        acc += mask[(long long)bb * S + i] ? 1 : 0;
    red[threadIdx.x] = acc;
    __syncthreads();
    for (int off = 128; off > 0; off >>= 1) {
        if ((int)threadIdx.x < off) red[threadIdx.x] += red[threadIdx.x + off];
        __syncthreads();
    }
    if (threadIdx.x == 0) out[bb] = red[0];
}

// ---------------------------------------------------------------------------
// Kernel 2/5: fp32 WMMA GEMM, C[M,N] = A[M,K] @ B[K,N], row-major.
// Block: 128 threads = 4 waves. Block tile 128x64, wave tile 32x64.
// K staged in 64-chunks in fragment order via async global->LDS. Per k-step
// a wave issues 2 A-frag + 4 B-frag ds_load_b64 feeding 8 WMMAs.
// Requires M % 128 == 0, N % 64 == 0, K % 64 == 0 (true: 8192x{3072,1024}x1024)
// ---------------------------------------------------------------------------
__global__ __launch_bounds__(128) void gemm_f32_wmma(
    const float* __restrict__ A, const float* __restrict__ B,
    float* __restrict__ C, int M, int N, int K) {
    __shared__ __align__(16) float Asf[8192];   // 128x64 chunk, frag order (32 KB)
    __shared__ __align__(16) float Bsf[4096];   // 64x64 chunk, frag order (16 KB)
    const int tid  = threadIdx.x;
    const int lane = tid & 31;
    const int wave = tid >> 5;
    const int half = lane >> 4;
    const int l16  = lane & 15;
    const int m0 = blockIdx.x * 128;
    const int n0 = blockIdx.y * 64;

    v8f acc[2][4] = {};   // wave owns 2 M-tiles x 4 N-tiles

    for (int kc = 0; kc < K; kc += 64) {
        // ---- stage A chunk 128x64: async b64 pairs into frag order ----
#pragma unroll
        for (int i = 0; i < 16; ++i) {
            const int linear = tid + i * 128;    // 0..2047
            const int c4 = linear & 15;          // kk
            const int m  = linear >> 4;          // 0..127
            const float* src = A + (size_t)(m0 + m) * K + kc + c4 * 4;
            const int mt = m >> 4, ml = m & 15;
            async_ld_b64(lds_off(&Asf[AFRAG(mt, c4, 0, ml)]), src);      // k=4c4+0,1
            async_ld_b64(lds_off(&Asf[AFRAG(mt, c4, 1, ml)]), src + 2);  // k=4c4+2,3
        }
        // ---- stage B chunk 64x64: async b32 scatter into frag order ----
#pragma unroll
        for (int i = 0; i < 8; ++i) {
            const int linear = tid + i * 128;    // 0..1023
            const int c4 = linear & 15;
            const int k  = linear >> 4;          // 0..63
            const float* src = B + (size_t)(kc + k) * N + n0 + c4 * 4;
            const int kk = k >> 2, hb = (k >> 1) & 1, sub = k & 1;
            const int j = c4 >> 2, nb = (c4 & 3) * 4;
            async_ld_b32(lds_off(&Bsf[BFRAG(kk, j, hb, nb + 0) + sub]), src + 0);
            async_ld_b32(lds_off(&Bsf[BFRAG(kk, j, hb, nb + 1) + sub]), src + 1);
            async_ld_b32(lds_off(&Bsf[BFRAG(kk, j, hb, nb + 2) + sub]), src + 2);
            async_ld_b32(lds_off(&Bsf[BFRAG(kk, j, hb, nb + 3) + sub]), src + 3);
        }
        wait_async0();
        __syncthreads();

#pragma unroll
        for (int kk = 0; kk < 16; ++kk) {
            const v2f a0 = *(const v2f*)&Asf[AFRAG(wave * 2 + 0, kk, half, l16)];
            const v2f a1 = *(const v2f*)&Asf[AFRAG(wave * 2 + 1, kk, half, l16)];
#pragma unroll
            for (int j = 0; j < 4; ++j) {
                const v2f b = *(const v2f*)&Bsf[BFRAG(kk, j, half, l16)];
                acc[0][j] = __builtin_amdgcn_wmma_f32_16x16x4_f32(
                    false, a0, false, b, (short)0, acc[0][j], false, false);
                acc[1][j] = __builtin_amdgcn_wmma_f32_16x16x4_f32(
                    false, a1, false, b, (short)0, acc[1][j], false, false);
            }
        }
        __syncthreads();
    }

    // C/D layout: vgpr i -> row i (lanes 0-15) or i+8 (lanes 16-31), col=l16.
#pragma unroll
    for (int mt = 0; mt < 2; ++mt)
#pragma unroll
        for (int j = 0; j < 4; ++j)
#pragma unroll
            for (int i = 0; i < 8; ++i) {
                const int row = m0 + wave * 32 + mt * 16 + i + half * 8;
                const int col = n0 + j * 16 + l16;
                C[(size_t)row * N + col] = acc[mt][j][i];
            }
}

// ---------------------------------------------------------------------------
// Kernel 3: RoPE + RMSNorm (+ per-dim softplus scale on q), in-place on the
// q and k slices of the qkv buffer. One wave32 per (b,s,head) 64-wide row;
// lane owns the rotation pair (x[lane], x[lane+32]).  HD must be 64.
// ---------------------------------------------------------------------------
__global__ __launch_bounds__(256) void rope_rms_kernel(
    float* __restrict__ qkv, const int* __restrict__ num_masked,
    const float* __restrict__ q_ln, const float* __restrict__ k_ln,
    const float* __restrict__ per_dim_scale,
    int B, int S, int H, int HD) {
    const int lane = threadIdx.x & 31;
    const int wave = threadIdx.x >> 5;
    const long long bsh = (long long)B * S * H;
    const long long row = (long long)blockIdx.x * 8 + wave;   // wave-uniform
    if (row >= 2 * bsh) return;

    const int isK = row >= bsh;
    const long long r = isK ? row - bsh : row;
    const int hh = (int)(r % H);
    const long long bs = r / H;
    const int ss = (int)(bs % S);
    const int bb = (int)(bs / S);
    const int d  = H * HD;
    float* base = qkv + ((long long)bb * S + ss) * (3 * d)
                  + (isK ? d : 0) + (long long)hh * HD;

    float x1 = base[lane];
    float x2 = base[lane + 32];

    // RoPE: timescale = 10000^(2i/d), i = lane, d = 64.
    const float pos = (float)ss - (float)num_masked[bb];
    const float ts  = __powf(10000.0f, (float)lane * (1.0f / 32.0f));
    const float ang = pos / ts;
    float sn, cs;
    __sincosf(ang, &sn, &cs);
    float r1 = x1 * cs - x2 * sn;
    float r2 = x2 * cs + x1 * sn;

    // RMSNorm across the 64-wide head dim (wave-wide reduction).
    float sq = r1 * r1 + r2 * r2;
#pragma unroll
    for (int m = 1; m < 32; m <<= 1) sq += __shfl_xor(sq, m, 32);
    const float inv = rsqrtf(sq * (1.0f / 64.0f) + RMS_EPS);
    const float* ln = isK ? k_ln : q_ln;
    r1 *= inv * ln[lane];
    r2 *= inv * ln[lane + 32];

    if (!isK) {
        // q *= LOG2_E / sqrt(hd) * softplus(per_dim_scale)
        const float p1 = per_dim_scale[lane];
        const float p2 = per_dim_scale[lane + 32];
        const float sp1 = p1 > 0.f ? p1 + log1pf(__expf(-p1)) : log1pf(__expf(p1));
        const float sp2 = p2 > 0.f ? p2 + log1pf(__expf(-p2)) : log1pf(__expf(p2));
        const float cf = LOG2_E * 0.125f;   // 1/sqrt(64) = 0.125
        r1 *= cf * sp1;
        r2 *= cf * sp2;
    }
    base[lane]      = r1;
    base[lane + 32] = r2;
}

// ---------------------------------------------------------------------------
// Kernel 4: causal flash attention with f32 WMMA.
// Grid: (b*h, s/64). Block: 128 threads = 4 waves; each wave owns 16 q rows,
// the block shares a 16x64 K tile (row-major, pad 68 -> b64 frag reads are
// already conflict-free) and a 16x64 V tile in fragment order, both staged
// with async global->LDS. P is staged per-wave in fragment order so the P@V
// loop is pure ds_load_b64 -> WMMA. The K/V row after each staged row is prefetched during
// staging. Wave-level causal skip is wave-uniform -> EXEC all-ones for WMMA.
// ---------------------------------------------------------------------------
__global__ __launch_bounds__(128) void flash_attn_kernel(
    const float* __restrict__ qkv, const int* __restrict__ num_masked_arr,
    float* __restrict__ ctx, int B, int S, int H, int HD) {
    __shared__ __align__(16) float Ks[16][GSTRIDE];
    __shared__ __align__(16) float Vsf[1024];      // 16x64 V, frag order (4 KB)
    __shared__ __align__(16) float Psf[4][256];    // per-wave 16x16 P, frag order

    const int tid  = threadIdx.x;
    const int lane = tid & 31;
    const int wave = tid >> 5;
    const int half = lane >> 4;
    const int l16  = lane & 15;
    const int bb = blockIdx.x / H;
    const int hh = blockIdx.x % H;
    const int q0  = blockIdx.y * 64;
    const int qw0 = q0 + wave * 16;
    const int d  = H * HD;
    const int d3 = 3 * d;
    const int nm = num_masked_arr[bb];

    // Preload Q A-fragments for all 16 k-steps (32 VGPRs/lane).
    float qa0[16], qa1[16];
    {
        const float* Qb = qkv + ((long long)bb * S + (qw0 + l16)) * d3
                          + (long long)hh * HD;
#pragma unroll
        for (int kk = 0; kk < 16; ++kk) {
            qa0[kk] = Qb[kk * 4 + half * 2 + 0];
            qa1[kk] = Qb[kk * 4 + half * 2 + 1];
        }
    }

    v8f o_acc[4] = {};
    float run_max[8], run_sum[8];
#pragma unroll
    for (int i = 0; i < 8; ++i) { run_max[i] = -1e30f; run_sum[i] = 0.f; }

    const int kt_lo = nm >> 4;
    const int kt_hi = (q0 + 63) >> 4;
    for (int kt = kt_lo; kt <= kt_hi; ++kt) {
        // Async stage of 16x64 K (row-major) and V (frag-order scatter).
#pragma unroll
        for (int i = 0; i < 2; ++i) {
            const int linear = tid + i * 128;   // 0..255
            const int r  = linear >> 4;
            const int c4 = linear & 15;
            const float* Kb = qkv + ((long long)bb * S + (kt * 16 + r)) * d3
                              + d + (long long)hh * HD;
            async_ld_b128(lds_off(&Ks[r][c4 * 4]), Kb + c4 * 4);
            const float* Vb = Kb + d + c4 * 4;
            const int kkv = r >> 2, hv = (r >> 1) & 1, sub = r & 1;
            const int j = c4 >> 2, nb = (c4 & 3) * 4;
            async_ld_b32(lds_off(&Vsf[BFRAG(kkv, j, hv, nb + 0) + sub]), Vb + 0);
            async_ld_b32(lds_off(&Vsf[BFRAG(kkv, j, hv, nb + 1) + sub]), Vb + 1);
            async_ld_b32(lds_off(&Vsf[BFRAG(kkv, j, hv, nb + 2) + sub]), Vb + 2);
            async_ld_b32(lds_off(&Vsf[BFRAG(kkv, j, hv, nb + 3) + sub]), Vb + 3);
            if (kt < kt_hi) {   // pull K/V row kt*16+r+1 toward L2/WGP$ while we compute
                __builtin_prefetch(Kb + d3 + c4 * 4, 0, 3);
                __builtin_prefetch(Kb + d3 + d + c4 * 4, 0, 3);
            }
        }
        wait_async0();
        __syncthreads();

        if (kt * 16 <= qw0 + 15) {   // wave-uniform causal extent
            // ---- S = Q @ K^T : 16 x f32 WMMA over hd=64 ----
            v8f s_acc = {};
#pragma unroll
            for (int kk = 0; kk < 16; ++kk) {
                v2f a; a.x = qa0[kk]; a.y = qa1[kk];
                // B[k][n] = K[row=n][dim=k]: b64 at row l16 (conflict-free)
                const v2f b = *(const v2f*)&Ks[l16][kk * 4 + half * 2];
                s_acc = __builtin_amdgcn_wmma_f32_16x16x4_f32(
                    false, a, false, b, (short)0, s_acc, false, false);
            }

            // ---- mask + online softmax (row lives on one 16-lane half) ----
            float p[8];
#pragma unroll
            for (int i = 0; i < 8; ++i) {
                const int qg = qw0 + i + half * 8;
                const int kg = kt * 16 + l16;
                float sv = s_acc[i];
                if (kg > qg || kg < nm) sv = -1e30f;
                float mx = sv;
                mx = fmaxf(mx, __shfl_xor(mx, 1, 32));
                mx = fmaxf(mx, __shfl_xor(mx, 2, 32));
                mx = fmaxf(mx, __shfl_xor(mx, 4, 32));
                mx = fmaxf(mx, __shfl_xor(mx, 8, 32));
                const float nmax  = fmaxf(run_max[i], mx);
                const float scale = __expf(run_max[i] - nmax);
                float pv = __expf(sv - nmax);
                if (sv <= -1e29f) pv = 0.f;   // keep fully-masked rows clean
                float sum = pv;
                sum += __shfl_xor(sum, 1, 32);
                sum += __shfl_xor(sum, 2, 32);
                sum += __shfl_xor(sum, 4, 32);
                sum += __shfl_xor(sum, 8, 32);
                run_sum[i] = run_sum[i] * scale + sum;
                run_max[i] = nmax;
                p[i] = pv;
#pragma unroll
                for (int j = 0; j < 4; ++j) o_acc[j][i] *= scale;
            }

            // Stage P into per-wave frag-order LDS:
            //   element (m = i + 8*half, n = l16):
            //   addr = ((n>>2)*16 + m)*4 + ((n>>1)&1)*2 + (n&1)
#pragma unroll
            for (int i = 0; i < 8; ++i)
                Psf[wave][((l16 >> 2) * 16 + i + half * 8) * 4
                          + ((l16 >> 1) & 1) * 2 + (l16 & 1)] = p[i];
            // (same-wave LDS RAW: hardware DScnt ordering, no barrier needed)

            // ---- O += P @ V : 16 x f32 WMMA (K=16, N=64) ----
#pragma unroll
            for (int kk = 0; kk < 4; ++kk) {
                const v2f a =
                    *(const v2f*)&Psf[wave][(kk * 16 + l16) * 4 + half * 2];
#pragma unroll
                for (int j = 0; j < 4; ++j) {
                    const v2f b = *(const v2f*)&Vsf[BFRAG(kk, j, half, l16)];
                    o_acc[j] = __builtin_amdgcn_wmma_f32_16x16x4_f32(
                        false, a, false, b, (short)0, o_acc[j], false, false);
                }
            }
        }
        __syncthreads();
    }

    // Epilogue: normalize and store ctx[b, q, h, hd].
#pragma unroll
    for (int i = 0; i < 8; ++i) {
        const float inv = run_sum[i] > 0.f ? 1.0f / run_sum[i] : 0.f;
        const int qg = qw0 + i + half * 8;
        float* Cb = ctx + ((long long)bb * S + qg) * d + (long long)hh * HD;
#pragma unroll
        for (int j = 0; j < 4; ++j) Cb[j * 16 + l16] = o_acc[j][i] * inv;
    }
}

// ---------------------------------------------------------------------------
// Launch: mask-sum -> QKV GEMM -> RoPE/RMSNorm -> flash attention -> out GEMM
// Workspace: qkv (b*s*3d f32) | ctx (b*s*d f32) | num_masked (b ints)
//            = 96 MB + 32 MB + 16 B
// ---------------------------------------------------------------------------
extern "C" void kernel_launch(void* const* d_in, const int* in_sizes, int n_in,
                              void* d_out, int out_size, void* d_ws, size_t ws_size,
                              hipStream_t stream) {
    const float*         inputs_q   = (const float*)d_in[0];
    const unsigned char* patch_mask = (const unsigned char*)d_in[1];  // bool
    const float*         w_qkv      = (const float*)d_in[2];
    const float*         w_out      = (const float*)d_in[3];
    const float*         q_ln       = (const float*)d_in[4];
    const float*         k_ln       = (const float*)d_in[5];
    const float*         pds        = (const float*)d_in[6];
    float*               out        = (float*)d_out;

    const int B = 4, S = 2048, H = 16, HD = 64;   // reference setup_inputs()
    const int D = H * HD;                         // 1024

    float* qkv = (float*)d_ws;                              // B*S*3D
    float* ctx = qkv + (size_t)B * S * 3 * D;               // B*S*D
    int*   nmk = (int*)(ctx + (size_t)B * S * D);           // B

    mask_sum_kernel<<<B, 256, 0, stream>>>(patch_mask, nmk, S);

    gemm_f32_wmma<<<dim3((B * S) / 128, (3 * D) / 64), 128, 0, stream>>>(
        inputs_q, w_qkv, qkv, B * S, 3 * D, D);

    const long long rows = 2LL * B * S * H;   // q rows + k rows
    rope_rms_kernel<<<(int)(rows / 8), 256, 0, stream>>>(
        qkv, nmk, q_ln, k_ln, pds, B, S, H, HD);

    flash_attn_kernel<<<dim3(B * H, S / 64), 128, 0, stream>>>(
        qkv, nmk, ctx, B, S, H, HD);

    gemm_f32_wmma<<<dim3((B * S) / 128, D / 64), 128, 0, stream>>>(
        ctx, w_out, out, B * S, D, D);
}